// RGCNLayer_15204184228257
// MI455X (gfx1250) — compile-verified
//
#include <hip/hip_runtime.h>

typedef float v2f __attribute__((ext_vector_type(2)));
typedef float v8f __attribute__((ext_vector_type(8)));

#define D_DIM   128
#define CHUNK   512
#define K2_THR  128
#define NWAVES  (K2_THR / 32)
#define A_STR   132   // padded A-tile row stride in dwords (conflict-free LDS)

// ---------------------------------------------------------------------------
// Edge kernel: per 512-edge chunk, counting-sort by relation, then WMMA tiles
// of 16 same-relation edges:  msg[16x128] = A[16x128] * W_r^T + b_r,
// scatter-add rows into acc[dst].
// ---------------------------------------------------------------------------
__global__ __launch_bounds__(K2_THR) void rgcn_edge_kernel(
    const float* __restrict__ x,      // [N,D]
    const int*   __restrict__ esrc,   // [E]
    const int*   __restrict__ edst,   // [E]
    const int*   __restrict__ etype,  // [E]
    const float* __restrict__ W,      // [R,D,D] (row-major, W[r][out][in])
    const float* __restrict__ bias,   // [R,D]
    float*       __restrict__ acc,    // [N,D] accumulator (pre-loaded with x)
    int E)
{
    __shared__ int   s_et[CHUNK];
    __shared__ int   s_eid[CHUNK];
    __shared__ int   s_cur[8];
    __shared__ int   s_start[9];
    __shared__ int   s_tile[9];
    __shared__ float s_A[NWAVES][16 * A_STR];

    const int tid       = threadIdx.x;
    const int chunkBase = blockIdx.x * CHUNK;
    const int nE        = min(CHUNK, E - chunkBase);

    // ---- stage edge types + histogram -------------------------------------
    if (tid < 8) s_cur[tid] = 0;
    __syncthreads();
    for (int i = tid; i < nE; i += K2_THR) {
        int r = etype[chunkBase + i] & 7;
        s_et[i] = r;
        atomicAdd(&s_cur[r], 1);
    }
    __syncthreads();
    // ---- prefix + stable serial scatter (deterministic tile membership) ---
    if (tid == 0) {
        int a = 0, t = 0;
        for (int r = 0; r < 8; ++r) {
            int c = s_cur[r];
            s_start[r] = a; s_tile[r] = t;
            s_cur[r]   = a;                 // reuse as cursor
            a += c; t += (c + 15) >> 4;
        }
        s_start[8] = a; s_tile[8] = t;
        for (int i = 0; i < nE; ++i) {
            int r = s_et[i];
            s_eid[s_cur[r]++] = chunkBase + i;
        }
    }
    __syncthreads();

    const int wave = tid >> 5;
    const int lane = tid & 31;
    const int hi   = lane >> 4;      // K-half selector (ISA A/B lane layout)
    const int l15  = lane & 15;
    const int nTiles = s_tile[8];
    float* Aw = &s_A[wave][0];

    for (int t = wave; t < nTiles; t += NWAVES) {
        int r = 0;
        while (s_tile[r + 1] <= t) ++r;                 // uniform per wave
        const int base = s_start[r] + ((t - s_tile[r]) << 4);
        const int cnt  = min(16, s_start[r + 1] - base);

        // per-lane edge for row m = l15 (lanes 16..31 duplicate 0..15)
        const int eid  = s_eid[base + min(l15, cnt - 1)];
        const int srcN = esrc[eid];
        const int dstN = edst[eid];

        // ---- gather A tile: 16 rows x 128 f32, one float4 per lane/row ----
        for (int row = 0; row < 16; ++row) {
            const int sr = __shfl(srcN, row, 32);
            const float4 v = *(const float4*)(x + (size_t)sr * D_DIM + lane * 4);
            *(float4*)(Aw + row * A_STR + lane * 4) = v;
        }
        // cross-lane LDS RAW within the wave: drain DS before fragment reads
        asm volatile("s_wait_dscnt 0" ::: "memory");

        // ---- 8 output column tiles, K=128 via 32 chained f32 WMMAs --------
        const float* Ar = Aw + l15 * A_STR + hi * 2;    // A[m=l15][k + hi*2]
        for (int n0 = 0; n0 < D_DIM; n0 += 16) {
            v8f c = {0.f, 0.f, 0.f, 0.f, 0.f, 0.f, 0.f, 0.f};
            // B[k][n] = W[r][n][k]; lane holds n = l15, k-pair = k0 + hi*2
            const float* Wr = W + ((size_t)r * D_DIM + n0 + l15) * D_DIM + hi * 2;
            #pragma unroll
            for (int k = 0; k < D_DIM; k += 4) {
                v2f a  = *(const v2f*)(Ar + k);
                v2f bb = *(const v2f*)(Wr + k);
                c = __builtin_amdgcn_wmma_f32_16x16x4_f32(
                        false, a, false, bb, (short)0, c, false, false);
            }
            const float bv = bias[r * D_DIM + n0 + l15]; // per-column bias
            // ---- scatter: lane holds C[M = j + 8*hi][N = n0 + l15] --------
            #pragma unroll
            for (int j = 0; j < 8; ++j) {
                const int M = j + hi * 8;
                const int d = __shfl(dstN, M, 32);
                if (M < cnt)
                    unsafeAtomicAdd(acc + (size_t)d * D_DIM + n0 + l15, c[j] + bv);
            }
        }
    }
}

// ---------------------------------------------------------------------------
// LayerNorm: one wave32 per row (4 f32 per lane), in-place on acc.
// ---------------------------------------------------------------------------
__global__ __launch_bounds__(256) void rgcn_ln_kernel(
    float* __restrict__ acc,
    const float* __restrict__ gamma,
    const float* __restrict__ beta,
    int N)
{
    const int wave = threadIdx.x >> 5;
    const int lane = threadIdx.x & 31;
    const int row  = blockIdx.x * 8 + wave;
    if (row >= N) return;

    float* p = acc + (size_t)row * D_DIM + lane * 4;
    float4 v = *(const float4*)p;
    float s  = v.x + v.y + v.z + v.w;
    float sq = v.x * v.x + v.y * v.y + v.z * v.z + v.w * v.w;
    #pragma unroll
    for (int off = 16; off > 0; off >>= 1) {
        s  += __shfl_xor(s,  off, 32);
        sq += __shfl_xor(sq, off, 32);
    }
    const float mu  = s * (1.0f / D_DIM);
    const float var = sq * (1.0f / D_DIM) - mu * mu;
    const float rs  = rsqrtf(var + 1e-5f);
    const float4 g  = *(const float4*)(gamma + lane * 4);
    const float4 be = *(const float4*)(beta  + lane * 4);
    float4 o;
    o.x = (v.x - mu) * rs * g.x + be.x;
    o.y = (v.y - mu) * rs * g.y + be.y;
    o.z = (v.z - mu) * rs * g.z + be.z;
    o.w = (v.w - mu) * rs * g.w + be.w;
    *(float4*)p = o;
}

// ---------------------------------------------------------------------------
extern "C" void kernel_launch(void* const* d_in, const int* in_sizes, int n_in,
                              void* d_out, int out_size, void* d_ws, size_t ws_size,
                              hipStream_t stream)
{
    const float* x     = (const float*)d_in[0];   // entity_emb [N,128]
    const int*   eidx  = (const int*)  d_in[1];   // edge_index [2,E]
    const int*   etype = (const int*)  d_in[2];   // edge_type  [E]
    const float* W     = (const float*)d_in[3];   // [R,128,128]
    const float* b     = (const float*)d_in[4];   // [R,128]
    const float* gamma = (const float*)d_in[5];   // [128]
    const float* beta  = (const float*)d_in[6];   // [128]

    const int N = in_sizes[0] / D_DIM;
    const int E = in_sizes[1] / 2;

    float* acc = (float*)d_out;

    // residual init: acc = entity_emb (D2D copy is graph-capture safe)
    hipMemcpyAsync(acc, x, (size_t)N * D_DIM * sizeof(float),
                   hipMemcpyDeviceToDevice, stream);

    const int nChunks = (E + CHUNK - 1) / CHUNK;
    rgcn_edge_kernel<<<nChunks, K2_THR, 0, stream>>>(
        x, eidx, eidx + E, etype, W, b, acc, E);

    rgcn_ln_kernel<<<(N + 7) / 8, 256, 0, stream>>>(acc, gamma, beta, N);
}